// MultiSpectralDCTLayer_47141561041431
// MI455X (gfx1250) — compile-verified
//
#include <hip/hip_runtime.h>

// CDNA5 f32 WMMA operand vectors (wave32): A/B = 2 VGPRs, C/D = 8 VGPRs.
typedef __attribute__((ext_vector_type(2))) float v2f;
typedef __attribute__((ext_vector_type(8))) float v8f;

#define SIZE 112
#define LSTR 113                 // LDS row stride (odd -> conflict-free strided reads)
#define NT 7                     // 7x7 tiles of 16x16
#define NTILES 49
#define SLICE (SIZE * SIZE)

// ---------------------------------------------------------------------------
// Kernel A: fill the entire output with 1e-8 (the masked value). 205 MB of
// stores -> this dominates runtime (~9us at 23.3 TB/s HBM). float4 stores.
// ---------------------------------------------------------------------------
__global__ void dct_fill_kernel(float4* __restrict__ out, int n4) {
    const float4 v = make_float4(1e-8f, 1e-8f, 1e-8f, 1e-8f);
    int stride = gridDim.x * blockDim.x;
    for (int i = blockIdx.x * blockDim.x + threadIdx.x; i < n4; i += stride)
        out[i] = v;
}

// ---------------------------------------------------------------------------
// Kernel B: for the only live channels (c in {0,1}), compute
//   Y = W * X * W^T   via V_WMMA_F32_16X16X4_F32 (exact f32 math),
// then store with the integer-exact band mask:
//   s = (i XOR 2) + (j XOR 2); keep iff (c==0 ? s<196 : s>=196).
// One workgroup (256 thr = 8 wave32) per (b,c) slice. W, X, T staged in LDS.
// ---------------------------------------------------------------------------
__global__ __launch_bounds__(256) void dct_wmma_kernel(const float* __restrict__ x,
                                                       const float* __restrict__ Wg,
                                                       float* __restrict__ out) {
    extern __shared__ float lds[];
    float* sW = lds;                    // 112 x 113
    float* sX = sW + SIZE * LSTR;       // 112 x 113
    float* sT = sX + SIZE * LSTR;       // 112 x 113 (T = W * X)

    const int tid  = threadIdx.x;
    const int wave = tid >> 5;
    const int lane = tid & 31;
    const int n    = lane & 15;         // N column within tile / A-row M
    const int hi   = lane >> 4;         // half-wave select

    const int b = blockIdx.x >> 1;
    const int c = blockIdx.x & 1;
    const float* xslice = x + (size_t)(b * 64 + c) * SLICE;

    // Cooperative load of W and the X slice into LDS (padded rows).
    for (int idx = tid; idx < SLICE; idx += 256) {
        int r = idx / SIZE, col = idx - r * SIZE;
        sW[r * LSTR + col] = Wg[idx];
        sX[r * LSTR + col] = xslice[idx];
    }
    __syncthreads();

    // ---- Stage 1: T = W * X ------------------------------------------------
    for (int t = wave; t < NTILES; t += 8) {      // uniform per wave
        const int ti = t / NT, tj = t - ti * NT;
        v8f acc = {};
        const float* arow = sW + (ti * 16 + n) * LSTR;   // A row M = n of W-tile
        const int    jcol = tj * 16 + n;                  // B column N = n
        for (int k0 = 0; k0 < SIZE; k0 += 4) {
            const int ka = k0 + 2 * hi;
            v2f a;  a.x = arow[ka];                 a.y = arow[ka + 1];
            v2f bb; bb.x = sX[ka * LSTR + jcol];    bb.y = sX[(ka + 1) * LSTR + jcol];
            acc = __builtin_amdgcn_wmma_f32_16x16x4_f32(
                false, a, false, bb, (short)0, acc, false, false);
        }
#pragma unroll
        for (int r = 0; r < 8; ++r)                 // D layout: rows r + 8*hi
            sT[(ti * 16 + r + 8 * hi) * LSTR + jcol] = acc[r];
    }
    __syncthreads();

    // ---- Stage 2: Y = T * W^T  (B[k][n] = W[tj*16+n][k]) -------------------
    float* od = out + (size_t)(b * 64 + c) * SLICE;
    for (int t = wave; t < NTILES; t += 8) {
        const int ti = t / NT, tj = t - ti * NT;
        v8f acc = {};
        const float* arow = sT + (ti * 16 + n) * LSTR;   // A row of T
        const float* brow = sW + (tj * 16 + n) * LSTR;   // W row = W^T column
        for (int k0 = 0; k0 < SIZE; k0 += 4) {
            const int ka = k0 + 2 * hi;
            v2f a;  a.x = arow[ka];  a.y = arow[ka + 1];
            v2f bb; bb.x = brow[ka]; bb.y = brow[ka + 1];
            acc = __builtin_amdgcn_wmma_f32_16x16x4_f32(
                false, a, false, bb, (short)0, acc, false, false);
        }
        const int j  = tj * 16 + n;
        const int xj = j ^ 2;
#pragma unroll
        for (int r = 0; r < 8; ++r) {
            const int i = ti * 16 + r + 8 * hi;
            const int s = (i ^ 2) + xj;                       // integer-exact mask
            const bool keep = (c == 0) ? (s < 196) : (s >= 196);
            od[i * SIZE + j] = keep ? acc[r] : 1e-8f;
        }
    }
}

// ---------------------------------------------------------------------------
extern "C" void kernel_launch(void* const* d_in, const int* in_sizes, int n_in,
                              void* d_out, int out_size, void* d_ws, size_t ws_size,
                              hipStream_t stream) {
    (void)in_sizes; (void)n_in; (void)d_ws; (void)ws_size;
    const float* x = (const float*)d_in[0];   // [64,64,112,112] f32
    const float* W = (const float*)d_in[1];   // [112,112] f32
    float* out = (float*)d_out;

    // 1) Fill everything with the masked constant (store-bandwidth bound).
    const int n4 = out_size / 4;
    dct_fill_kernel<<<4096, 256, 0, stream>>>((float4*)out, n4);

    // 2) Compute the 128 live slices (b in [0,64), c in {0,1}) with f32 WMMA.
    const size_t smem = (size_t)3 * SIZE * LSTR * sizeof(float); // ~148 KB < 320 KB
    dct_wmma_kernel<<<64 * 2, 256, smem, stream>>>(x, W, out);
}